// Repro_78726750535798
// MI455X (gfx1250) — compile-verified
//
#include <hip/hip_runtime.h>
#include <stdint.h>

typedef __attribute__((ext_vector_type(2))) float v2f;
typedef __attribute__((ext_vector_type(8))) float v8f;

#define IN_SIZE  500
#define OUT_SIZE 256
#define KSPAN    32   // max input span feeding a 16-wide output tile (29.3 + 2 taps)
#define PW       36   // patch width: KSPAN + (c_align slack) = 32 + 4; 144B rows (16B aligned)

// coord/idx/weight math identical to the reference (float32 throughout)
__device__ __forceinline__ void bilin_iw(int j, int& i0c, int& i1c, float& w0, float& w1) {
    const float scale = (float)IN_SIZE / (float)OUT_SIZE; // 1.953125 exact
    float coord = fmaxf(((float)j + 0.5f) * scale - 0.5f, 0.0f);
    int i0 = (int)coord;                 // floor, coord >= 0
    float frac = coord - (float)i0;      // in [0,1)
    i0c = min(i0, IN_SIZE - 1);
    i1c = min(i0 + 1, IN_SIZE - 1);
    w0 = 1.0f - frac;
    w1 = frac;
}

__device__ __forceinline__ int span_base(int j0) {
    const float scale = (float)IN_SIZE / (float)OUT_SIZE;
    float coord = fmaxf(((float)j0 + 0.5f) * scale - 0.5f, 0.0f);
    return (int)coord;                   // unclamped floor: tile's first tap index
}

__global__ __launch_bounds__(32)
void resize_bilinear_wmma(const float* __restrict__ x, float* __restrict__ out) {
    __shared__ float P[KSPAN][PW];       // input patch (rows row_lo.., cols c_align..)
    __shared__ float T[KSPAN][17];       // horizontal result, padded stride

    const int lane = threadIdx.x;        // 0..31, one wave per block
    const int half = lane >> 4;          // lane half selects K offset {0,2}
    const int lm   = lane & 15;          // M (A) / N (B,C,D) index
    const int kofs = half * 2;

    const int xt = blockIdx.x;           // output col tile (16 tiles)
    const int yt = blockIdx.y;           // output row tile
    const int bc = blockIdx.z;           // 0..191 (b*c)

    const float* img = x + (size_t)bc * IN_SIZE * IN_SIZE;

    // per-lane output column taps (same for both lane halves since lm matches)
    int xi0, xi1; float wx0, wx1;
    bilin_iw(xt * 16 + lm, xi0, xi1, wx0, wx1);
    // per-lane output row taps (for the vertical A matrix)
    int yi0, yi1; float wy0, wy1;
    bilin_iw(yt * 16 + lm, yi0, yi1, wy0, wy1);

    const int col_lo  = span_base(xt * 16);
    const int row_lo  = span_base(yt * 16);
    const int c_align = col_lo & ~3;     // 16B-aligned global column base
    const int coff    = col_lo - c_align; // 0..3 shift of valid data inside patch

    // ------------- Async-copy the 32x36 patch into LDS (9 b128 per lane) -----
    // Flat float4 index t = it*32 + lane over (32 rows x 9 float4/row).
    const uint32_t lds_base = (uint32_t)(uintptr_t)(&P[0][0]); // low 32b = LDS offset
#pragma unroll
    for (int it = 0; it < 9; ++it) {
        const int t   = it * 32 + lane;
        const int row = t / 9;
        const int j   = t - row * 9;            // float4 index within row
        const int grow = min(row_lo + row, IN_SIZE - 1);
        const int gcol = min(c_align + 4 * j, IN_SIZE - 4); // groups never straddle 499/500
        const uint32_t voff = (uint32_t)(grow * IN_SIZE + gcol) * 4u;
        const uint32_t ldsa = lds_base + (uint32_t)(row * PW + 4 * j) * 4u;
        asm volatile("global_load_async_to_lds_b128 %0, %1, %2"
                     :: "v"(ldsa), "v"(voff), "s"(img)
                     : "memory");
    }
    asm volatile("s_wait_asynccnt 0x0" ::: "memory");
    __syncthreads();

    // ---------------- Stage 1: horizontal  T(32x16) = P(32x32) * Bx(32x16) ---
    v8f t0 = {};   // input rows row_lo .. row_lo+15
    v8f t1 = {};   // input rows row_lo+16 .. row_lo+31
#pragma unroll
    for (int kk = 0; kk < 8; ++kk) {
        const int k0 = kk * 4 + kofs;
        const int c0 = col_lo + k0;
        const int c1 = c0 + 1;
        // B operand: weight of input col c for this lane's output col
        v2f b;
        b.x = (c0 == xi0 ? wx0 : 0.0f) + (c0 == xi1 ? wx1 : 0.0f);
        b.y = (c1 == xi0 ? wx0 : 0.0f) + (c1 == xi1 ? wx1 : 0.0f);

        v2f a0; a0.x = P[lm][coff + k0];      a0.y = P[lm][coff + k0 + 1];
        v2f a1; a1.x = P[16 + lm][coff + k0]; a1.y = P[16 + lm][coff + k0 + 1];

        t0 = __builtin_amdgcn_wmma_f32_16x16x4_f32(false, a0, false, b, (short)0, t0, false, false);
        t1 = __builtin_amdgcn_wmma_f32_16x16x4_f32(false, a1, false, b, (short)0, t1, false, false);
    }

    // ---------------- D-layout -> B-layout relayout through LDS --------------
    __syncthreads();
#pragma unroll
    for (int v = 0; v < 8; ++v) {
        T[v + half * 8][lm]      = t0[v];
        T[16 + v + half * 8][lm] = t1[v];
    }
    __syncthreads();

    // ---------------- Stage 2: vertical  out(16x16) = Wy(16x32) * T(32x16) ---
    v8f acc = {};
#pragma unroll
    for (int kk = 0; kk < 8; ++kk) {
        const int k0 = kk * 4 + kofs;
        const int r0 = row_lo + k0;
        const int r1 = r0 + 1;
        // A operand: weight of input row r for this lane's output row
        v2f a;
        a.x = (r0 == yi0 ? wy0 : 0.0f) + (r0 == yi1 ? wy1 : 0.0f);
        a.y = (r1 == yi0 ? wy0 : 0.0f) + (r1 == yi1 ? wy1 : 0.0f);
        // B operand: T rows k0, k0+1 at this lane's column
        v2f b;
        b.x = T[k0][lm];
        b.y = T[k0 + 1][lm];
        acc = __builtin_amdgcn_wmma_f32_16x16x4_f32(false, a, false, b, (short)0, acc, false, false);
    }

    // ---------------- Epilogue: round-to-nearest-even, store as float --------
    float* op = out + (size_t)bc * OUT_SIZE * OUT_SIZE + (size_t)(xt * 16 + lm);
#pragma unroll
    for (int v = 0; v < 8; ++v) {
        const int row = yt * 16 + v + half * 8;
        float val = rintf(acc[v]);                 // matches jnp.round (half-even)
        val = fminf(fmaxf(val, 0.0f), 255.0f);     // uint8 range (convex combo anyway)
        op[(size_t)row * OUT_SIZE] = val;
    }
}

extern "C" void kernel_launch(void* const* d_in, const int* in_sizes, int n_in,
                              void* d_out, int out_size, void* d_ws, size_t ws_size,
                              hipStream_t stream) {
    (void)in_sizes; (void)n_in; (void)d_ws; (void)ws_size; (void)out_size;
    const float* x = (const float*)d_in[0];
    float* out = (float*)d_out;
    dim3 grid(OUT_SIZE / 16, OUT_SIZE / 16, 64 * 3);   // 16 x 16 x 192 tiles
    resize_bilinear_wmma<<<grid, dim3(32), 0, stream>>>(x, out);
}